// ConvLRUBlock_81716047773943
// MI455X (gfx1250) — compile-verified
//
#include <hip/hip_runtime.h>
#include <cstdint>

// Problem constants (match reference)
constexpr int Bc = 4, Lc = 8, Cc = 32, Hc = 128, Wc = 256;
constexpr int BL = Bc * Lc;          // 32 images per channel
constexpr int FR = 64;               // selected freq rows (0..31, 96..127)
constexpr int FC = 32;               // selected freq cols
constexpr int NMODE = FR * FC;       // 2048

typedef __attribute__((ext_vector_type(2))) float v2f;
typedef __attribute__((ext_vector_type(8))) float v8f;
typedef __attribute__((ext_vector_type(4))) unsigned int u32x4;
typedef __attribute__((ext_vector_type(4))) int i32x4;
typedef __attribute__((ext_vector_type(8))) int i32x8;

// ---------------------------------------------------------------------------
// Wave-level 16x16 output tile: acc += A[16xK] * B[Kx16], both row-major.
// Fragment layout per CDNA5 ISA 7.12.2 (f32 16x16x4).
// ---------------------------------------------------------------------------
__device__ inline v8f wmma_loop(const float* __restrict__ A, int lda,
                                const float* __restrict__ B, int ldb,
                                int K, v8f acc) {
  const int lane = threadIdx.x & 31;
  const int half = lane >> 4;
  const int l    = lane & 15;
  for (int kk = 0; kk < K; kk += 4) {
    v2f a, b;
    a.x = A[l * lda + kk + 2 * half + 0];
    a.y = A[l * lda + kk + 2 * half + 1];
    b.x = B[(kk + 2 * half + 0) * ldb + l];
    b.y = B[(kk + 2 * half + 1) * ldb + l];
    acc = __builtin_amdgcn_wmma_f32_16x16x4_f32(
        false, a, false, b, (short)0, acc, false, false);
  }
  return acc;
}

__device__ inline void store_tile(float* __restrict__ Cb, int ldc, v8f c) {
  const int lane = threadIdx.x & 31;
  const int half = lane >> 4;
  const int l    = lane & 15;
#pragma unroll
  for (int r = 0; r < 8; ++r) Cb[(half * 8 + r) * ldc + l] = c[r];
}

// ---------------------------------------------------------------------------
// Tensor Data Mover: 1-D contiguous block of `nelem` f32 between global & LDS.
// D# per CDNA5 ISA ch.8: group0 = {count=1, lds_addr, global_addr, type=2},
// group1 = {data_size=4B, tensor_dim0=nelem, tensor_dim1=1, tile_dim0=nelem,
//           tile_dim1=1, tensor_dim0_stride=nelem}. Groups 2/3 zero (<=2D).
// amdgpu-toolchain builtin arity: 6 args (g0, g1, g2, g3, g4, cpol).
// ---------------------------------------------------------------------------
__device__ inline void tdm_copy_1d(bool store, const void* gptr,
                                   unsigned lds_off, int nelem) {
  unsigned long long ga = (unsigned long long)(uintptr_t)gptr;
  u32x4 g0;
  g0.x = 1u;                                   // count = 1 (valid descriptor)
  g0.y = lds_off;                              // LDS byte address
  g0.z = (unsigned)ga;                         // global addr [31:0]
  g0.w = (unsigned)((ga >> 32) & 0x01FFFFFFull) | 0x80000000u;  // [56:32]|type=2
  i32x8 g1;
  g1[0] = 0x00020000;                          // data_size = 2 -> 4 bytes
  g1[1] = (int)((unsigned)(nelem & 0xFFFF) << 16);   // tensor_dim0 [15:0]
  g1[2] = (int)(((unsigned)nelem >> 16) | (1u << 16)); // dim0[31:16] | dim1=1
  g1[3] = (int)((unsigned)(nelem & 0xFFFF) << 16);   // tile_dim0
  g1[4] = 1;                                   // tile_dim1 = 1
  g1[5] = nelem;                               // tensor_dim0_stride [31:0]
  g1[6] = 0;
  g1[7] = 0;
  i32x4 z4 = {0, 0, 0, 0};
  i32x8 z8 = {0, 0, 0, 0, 0, 0, 0, 0};
  if (store)
    __builtin_amdgcn_tensor_store_from_lds(g0, g1, z4, z4, z8, 0);
  else
    __builtin_amdgcn_tensor_load_to_lds(g0, g1, z4, z4, z8, 0);
}

// ---------------------------------------------------------------------------
// Twiddle tables (built each launch; deterministic):
//  FW [256][64]   : forward W-DFT, cols interleaved (re,im) for j=0..31
//  TB [128][256]  : forward H-DFT realified: rows 2ii+q, cols 2h+p
//  GH [256][128]  : inverse H-DFT realified (1/H folded): rows 2h+p, cols 2ii+q
//  GW [64][256]   : Hermitian inverse W (1/W + factor-2 folded): rows 2j+p
// ---------------------------------------------------------------------------
__global__ void init_tw(float* FW, float* TB, float* GH, float* GW) {
  const int idx = blockIdx.x * blockDim.x + threadIdx.x;
  const float PI2 = 6.28318530717958647692f;
  if (idx < 256 * 64) {               // FW
    int w = idx >> 6, n = idx & 63, j = n >> 1, p = n & 1;
    float th = -PI2 * (float)((j * w) & 255) / 256.0f;
    FW[idx] = p ? __sinf(th) : __cosf(th);
  }
  if (idx < 128 * 256) {              // TB
    int row = idx >> 8, col = idx & 255;
    int ii = row >> 1, q = row & 1, h = col >> 1, p = col & 1;
    int r = (ii < 32) ? ii : ii + 64;
    float ph = -PI2 * (float)((r * h) & 127) / 128.0f;
    float tr = __cosf(ph), ti = __sinf(ph);
    TB[idx] = (q == 0) ? ((p == 0) ? tr : -ti) : ((p == 0) ? ti : tr);
  }
  if (idx < 256 * 128) {              // GH
    int row = idx >> 7, col = idx & 127;
    int h = row >> 1, p = row & 1, ii = col >> 1, q = col & 1;
    int r = (ii < 32) ? ii : ii + 64;
    float ph = PI2 * (float)((r * h) & 127) / 128.0f;
    float gr = __cosf(ph) * (1.0f / 128.0f), gi = __sinf(ph) * (1.0f / 128.0f);
    GH[idx] = (p == 0) ? ((q == 0) ? gr : -gi) : ((q == 0) ? gi : gr);
  }
  if (idx < 64 * 256) {               // GW
    int row = idx >> 8, w = idx & 255;
    int j = row >> 1, p = row & 1;
    float a = ((j == 0) ? 1.0f : 2.0f) * (1.0f / 256.0f);
    float al = PI2 * (float)((j * w) & 255) / 256.0f;
    GW[idx] = (p == 0) ? a * __cosf(al) : -a * __sinf(al);
  }
}

// ---------------------------------------------------------------------------
// Forward: per-image partial rFFT2 restricted to 64x32 modes.
// Image is DMA'd into LDS via TENSOR_LOAD_TO_LDS, then two WMMA GEMM stages.
// Stage-B epilogue (KF write or kf-multiply + scatter) runs in-fragment:
// C rows m, m+1 of a tile are c[r], c[r+1] of the same lane -> (re,im) local.
//  which==0: images are k's channels -> raw modes to KF[c][mode][2]
//  which==1: images are x -> multiply by KF, scatter mode-major into ZB
// ZB layout: [mode][2c+p][bl]  (contiguous 64x32 block per mode)
// ---------------------------------------------------------------------------
__global__ __launch_bounds__(256) void fwd_kernel(
    const float* __restrict__ img, const float* __restrict__ FW,
    const float* __restrict__ TB, float* __restrict__ KF,
    float* __restrict__ ZB, int which) {
  __shared__ float Img[Hc * Wc];      // 128 KB, filled by TDM
  __shared__ float Xstack[256 * 32];  // 32 KB, [2h+p][j]

  const int id = blockIdx.x;
  const int wave = threadIdx.x >> 5;
  const int lane = threadIdx.x & 31;
  const int half = lane >> 4;
  const int l    = lane & 15;

  // Async DMA of the whole image into LDS (one issue from wave 0).
  if (wave == 0) {
    tdm_copy_1d(false, img + (size_t)id * (Hc * Wc),
                (unsigned)(size_t)(void*)&Img[0], Hc * Wc);
    __builtin_amdgcn_s_wait_tensorcnt(0);
  }
  __syncthreads();

  // Stage A: Img[128x256] x FW[256x64]; scatter into Xstack layout.
  for (int t = wave; t < 32; t += 8) {
    int tm = t >> 2, tn = t & 3;
    v8f acc = {};
    acc = wmma_loop(Img + tm * 16 * Wc, Wc, FW + tn * 16, 64, 256, acc);
#pragma unroll
    for (int r = 0; r < 8; ++r) {
      int h  = tm * 16 + half * 8 + r;
      int cn = tn * 16 + l;
      Xstack[(2 * h + (cn & 1)) * 32 + (cn >> 1)] = acc[r];
    }
  }
  __syncthreads();

  // Stage B: TB[128x256] x Xstack[256x32]; epilogue straight from fragments.
  const int c  = id % Cc;
  const int bl = id / Cc;
  for (int t = wave; t < 16; t += 8) {
    int tm = t >> 1, tn = t & 1;
    __builtin_prefetch(TB + tm * 16 * 256 + l * 256, 0, 1);
    v8f acc = {};
    acc = wmma_loop(TB + tm * 16 * 256, 256, Xstack + tn * 16, 32, 256, acc);
    if (which == 0) {
#pragma unroll
      for (int r = 0; r < 8; ++r) {
        int r2 = tm * 16 + half * 8 + r;            // = 2*ii + p
        int ii = r2 >> 1, p = r2 & 1, j = tn * 16 + l;
        KF[(size_t)id * (NMODE * 2) + (ii * 32 + j) * 2 + p] = acc[r];
      }
    } else {
#pragma unroll
      for (int q = 0; q < 4; ++q) {
        int r2 = tm * 16 + half * 8 + 2 * q;        // even row -> re
        int ii = r2 >> 1, j = tn * 16 + l;
        float xr = acc[2 * q], xi = acc[2 * q + 1];
        size_t ko = (size_t)c * (NMODE * 2) + (ii * 32 + j) * 2;
        float kr = KF[ko], ki = KF[ko + 1];
        size_t base = ((size_t)(ii * 32 + j) * 64 + 2 * c) * 32 + bl;
        ZB[base]      = xr * kr - xi * ki;
        ZB[base + 32] = xr * ki + xi * kr;
      }
    }
  }
}

// ---------------------------------------------------------------------------
// Mode mix: one WG per (ii,j) mode. Realified complex channel mix:
//   Y[64x32] = Wbig[64x64] * Z[64x32]    (rows 2o+q, K = 2c+p, cols = bl)
// ---------------------------------------------------------------------------
__global__ __launch_bounds__(256) void mix_kernel(
    const float* __restrict__ w1r, const float* __restrict__ w1i,
    const float* __restrict__ w2r, const float* __restrict__ w2i,
    const float* __restrict__ ZB, float* __restrict__ YB) {
  __shared__ float Wb[64 * 64];
  __shared__ float Zin[64 * 32];
  const int mode = blockIdx.x;           // ii*32 + j
  const int ii = mode >> 5, j = mode & 31;
  const float* wr = (ii < 32) ? w1r : w2r;
  const float* wi = (ii < 32) ? w1i : w2i;
  const int i = ii & 31;

  for (int idx = threadIdx.x; idx < 64 * 32; idx += 256)
    Zin[idx] = ZB[(size_t)mode * 2048 + idx];
  for (int idx = threadIdx.x; idx < 32 * 32; idx += 256) {
    int o = idx >> 5, c = idx & 31;
    size_t off = (((size_t)c * 32 + o) * 32 + i) * 32 + j;  // w[c][o][i][j]
    float r = wr[off], m = wi[off];
    Wb[(2 * o) * 64 + 2 * c]         = r;
    Wb[(2 * o) * 64 + 2 * c + 1]     = -m;
    Wb[(2 * o + 1) * 64 + 2 * c]     = m;
    Wb[(2 * o + 1) * 64 + 2 * c + 1] = r;
  }
  __syncthreads();

  const int wave = threadIdx.x >> 5;     // 8 waves -> 8 tiles (4x2)
  int tm = wave >> 1, tn = wave & 1;
  v8f acc = {};
  acc = wmma_loop(Wb + tm * 16 * 64, 64, Zin + tn * 16, 32, 64, acc);
  store_tile(YB + (size_t)mode * 2048 + tm * 16 * 32 + tn * 16, 32, acc);
}

// ---------------------------------------------------------------------------
// Inverse: one WG per (bl,o) output image.
//   GEMM1: Yh2[256x32] = GH[256x128] * Ym[128x32]
//   repack -> Yhm[128][64]
//   GEMM2: Out[128x256] = Yhm[128x64] * GW[64x256]  (assembled in LDS)
//   TENSOR_STORE_FROM_LDS pushes the 128 KB image to global.
// LDS aliasing by liveness: Ym/Yh2 live only before Out is written.
// ---------------------------------------------------------------------------
__global__ __launch_bounds__(256) void inv_kernel(
    const float* __restrict__ YB, const float* __restrict__ GH,
    const float* __restrict__ GW, float* __restrict__ out) {
  __shared__ float S[40960];            // 160 KB total
  float* Yhm = S;                       // [128][64]  (8192)   lives into GEMM2
  float* Ym  = S + 8192;                // [128][32]  (4096)   dead after GEMM1
  float* Yh2 = S + 12288;               // [256][32]  (8192)   dead after repack
  float* Out = S + 8192;                // [128][256] (32768)  written in GEMM2

  const int id = blockIdx.x;            // bl*32 + o
  const int o = id & 31, bl = id >> 5;
  const int wave = threadIdx.x >> 5;
  const int lane = threadIdx.x & 31;
  const int l    = lane & 15;

  for (int idx = threadIdx.x; idx < FR * FC * 2; idx += 256) {
    int p = idx & 1, j = (idx >> 1) & 31, ii = idx >> 6;
    Ym[(2 * ii + p) * 32 + j] =
        YB[((size_t)(ii * 32 + j) * 64 + 2 * o + p) * 32 + bl];
  }
  __syncthreads();

  for (int t = wave; t < 32; t += 8) {        // GEMM1: 16x2 tiles
    int tm = t >> 1, tn = t & 1;
    v8f acc = {};
    acc = wmma_loop(GH + tm * 16 * 128, 128, Ym + tn * 16, 32, 128, acc);
    store_tile(&Yh2[tm * 16 * 32 + tn * 16], 32, acc);
  }
  __syncthreads();

  for (int idx = threadIdx.x; idx < 128 * 64; idx += 256) {
    int h = idx >> 6, n = idx & 63;
    Yhm[idx] = Yh2[(2 * h + (n & 1)) * 32 + (n >> 1)];
  }
  __syncthreads();

  for (int t = wave; t < 128; t += 8) {       // GEMM2: 8x16 tiles
    int tm = t >> 4, tn = t & 15;
    __builtin_prefetch(GW + tn * 16 + l, 0, 1);
    v8f acc = {};
    acc = wmma_loop(Yhm + tm * 16 * 64, 64, GW + tn * 16, 256, 64, acc);
    store_tile(Out + tm * 16 * 256 + tn * 16, 256, acc);
  }
  __syncthreads();

  if (wave == 0) {
    tdm_copy_1d(true, out + (size_t)id * (Hc * Wc),
                (unsigned)(size_t)(void*)&Out[0], Hc * Wc);
    __builtin_amdgcn_s_wait_tensorcnt(0);
  }
}

// ---------------------------------------------------------------------------
extern "C" void kernel_launch(void* const* d_in, const int* in_sizes, int n_in,
                              void* d_out, int out_size, void* d_ws,
                              size_t ws_size, hipStream_t stream) {
  (void)in_sizes; (void)n_in; (void)out_size; (void)ws_size;
  const float* x   = (const float*)d_in[0];
  const float* k   = (const float*)d_in[1];
  const float* w1r = (const float*)d_in[2];
  const float* w1i = (const float*)d_in[3];
  const float* w2r = (const float*)d_in[4];
  const float* w2i = (const float*)d_in[5];

  float* ws = (float*)d_ws;
  float* FW = ws;                    // 256*64    = 16384
  float* TB = ws + 16384;            // 128*256   = 32768
  float* GH = ws + 49152;            // 256*128   = 32768
  float* GW = ws + 81920;            // 64*256    = 16384
  float* KF = ws + 98304;            // 32*2048*2 = 131072
  float* ZB = ws + 229376;           // 2048*64*32 = 4194304
  float* YB = ws + 4423680;          // 2048*64*32 = 4194304

  init_tw<<<128, 256, 0, stream>>>(FW, TB, GH, GW);
  fwd_kernel<<<Cc, 256, 0, stream>>>(k, FW, TB, KF, ZB, 0);       // k modes
  fwd_kernel<<<BL * Cc, 256, 0, stream>>>(x, FW, TB, KF, ZB, 1);  // x modes * kf
  mix_kernel<<<NMODE, 256, 0, stream>>>(w1r, w1i, w2r, w2i, ZB, YB);
  inv_kernel<<<BL * Cc, 256, 0, stream>>>(YB, GH, GW, (float*)d_out);
}